// AttnReductionFusionEncoder_63299228008793
// MI455X (gfx1250) — compile-verified
//
#include <hip/hip_runtime.h>

typedef __attribute__((ext_vector_type(2))) float v2f;
typedef __attribute__((ext_vector_type(8))) float v8f;

#define BDIM 128
#define VDIM 1024
#define TDIM 1024
#define CDIM 1024

// e^{tanh(x)} with a NaN-safe tanh: tanh(x) = 1 - 2/(e^{2x}+1)
// (e2x -> inf gives tanh -> 1, e2x -> 0 gives tanh -> -1; no inf/inf)
__device__ __forceinline__ float exp_tanh(float x) {
    float e2x = __expf(2.0f * x);
    float th  = 1.0f - __fdividef(2.0f, e2x + 1.0f);
    return __expf(th);
}

// Pass 1 (column pass, softmax denominator lives along T):
//   D[b,v]  = sum_t exp(tanh(av*w_vis[t] + text[b,t]*w_text[v] + bias[v]))
//   ts[b,v] = (sum_t e * text[b,t]) / D[b,v]
//   vscaled[b,v] = visual[b,v] / D[b,v]        (consumed by pass 2)
__global__ __launch_bounds__(256) void k_colpass(
    const float* __restrict__ visual, const float* __restrict__ text,
    const float* __restrict__ w_vis,  const float* __restrict__ w_text,
    const float* __restrict__ bias,
    float* __restrict__ text_score, float* __restrict__ vscaled) {
    __shared__ float s_text[TDIM];
    __shared__ float s_wvis[TDIM];
    const int b   = blockIdx.y;
    const int tid = threadIdx.x;
    for (int i = tid; i < TDIM; i += 256) {
        s_text[i] = text[b * TDIM + i];
        s_wvis[i] = w_vis[i];
    }
    __syncthreads();
    const int v   = blockIdx.x * 256 + tid;
    const float av = visual[b * VDIM + v];
    const float wt = w_text[v];
    const float bb = bias[v];
    float D = 0.0f, N = 0.0f;
#pragma unroll 8
    for (int t = 0; t < TDIM; ++t) {
        float e = exp_tanh(fmaf(av, s_wvis[t], fmaf(s_text[t], wt, bb)));
        D += e;
        N = fmaf(e, s_text[t], N);
    }
    float rD = __fdividef(1.0f, D);
    text_score[b * VDIM + v] = N * rD;
    vscaled[b * VDIM + v]    = av * rD;
}

// Pass 2 (row pass): vs[b,t] = sum_v e[b,t,v] * vscaled[b,v]
__global__ __launch_bounds__(256) void k_rowpass(
    const float* __restrict__ visual, const float* __restrict__ text,
    const float* __restrict__ w_vis,  const float* __restrict__ w_text,
    const float* __restrict__ bias,   const float* __restrict__ vscaled,
    float* __restrict__ visual_score) {
    __shared__ float s_vis[VDIM];
    __shared__ float s_wtext[VDIM];
    __shared__ float s_bias[VDIM];
    __shared__ float s_vsc[VDIM];
    const int b   = blockIdx.y;
    const int tid = threadIdx.x;
    for (int i = tid; i < VDIM; i += 256) {
        s_vis[i]   = visual[b * VDIM + i];
        s_wtext[i] = w_text[i];
        s_bias[i]  = bias[i];
        s_vsc[i]   = vscaled[b * VDIM + i];
    }
    __syncthreads();
    const int t    = blockIdx.x * 256 + tid;
    const float ct = text[b * TDIM + t];
    const float wv = w_vis[t];
    float acc = 0.0f;
#pragma unroll 8
    for (int v = 0; v < VDIM; ++v) {
        float e = exp_tanh(fmaf(s_vis[v], wv, fmaf(ct, s_wtext[v], s_bias[v])));
        acc = fmaf(e, s_vsc[v], acc);
    }
    visual_score[b * TDIM + t] = acc;
}

// Pass 3: out[b,c] = relu( sum_t vs[b,t]*W_fv[c,t] + sum_v ts[b,v]*W_ft[c,v]
//                          + b_fv[c] + b_ft[c] )
// fp32 GEMM on the matrix pipe: V_WMMA_F32_16X16X4_F32, one wave per 16x16
// output tile. A fragment (16x4): lane m=lane%16, K pair at kb + 2*(lane/16).
// B fragment (4x16, B[k][n] = W[c=n, t=k]): same per-lane K pair from row n.
__global__ __launch_bounds__(128) void k_fusion_wmma(
    const float* __restrict__ vs, const float* __restrict__ ts,
    const float* __restrict__ W_fv, const float* __restrict__ b_fv,
    const float* __restrict__ W_ft, const float* __restrict__ b_ft,
    float* __restrict__ out) {
    const int lane = threadIdx.x & 31;
    const int wave = threadIdx.x >> 5;
    const int n0   = (blockIdx.x * 4 + wave) * 16;  // C tile
    const int m0   = blockIdx.y * 16;               // B (batch) tile
    const int mn   = lane & 15;
    const int koff = (lane >> 4) * 2;

    v8f acc = {0.f, 0.f, 0.f, 0.f, 0.f, 0.f, 0.f, 0.f};

    const float* Arow = vs   + (m0 + mn) * TDIM + koff;
    const float* Brow = W_fv + (n0 + mn) * TDIM + koff;
#pragma unroll 4
    for (int kb = 0; kb < TDIM; kb += 4) {
        v2f a  = *(const v2f*)(Arow + kb);
        v2f bm = *(const v2f*)(Brow + kb);
        acc = __builtin_amdgcn_wmma_f32_16x16x4_f32(
            false, a, false, bm, (short)0, acc, false, false);
    }

    Arow = ts   + (m0 + mn) * VDIM + koff;
    Brow = W_ft + (n0 + mn) * VDIM + koff;
#pragma unroll 4
    for (int kb = 0; kb < VDIM; kb += 4) {
        v2f a  = *(const v2f*)(Arow + kb);
        v2f bm = *(const v2f*)(Brow + kb);
        acc = __builtin_amdgcn_wmma_f32_16x16x4_f32(
            false, a, false, bm, (short)0, acc, false, false);
    }

    // C/D layout: VGPR r -> M = m0 + r + 8*(lane/16), N = n0 + lane%16
    const int n     = n0 + mn;
    const float bb  = b_fv[n] + b_ft[n];
    const int mhalf = (lane >> 4) * 8;
#pragma unroll
    for (int r = 0; r < 8; ++r) {
        float val = acc[r] + bb;
        out[(m0 + mhalf + r) * CDIM + n] = fmaxf(val, 0.0f);
    }
}

extern "C" void kernel_launch(void* const* d_in, const int* in_sizes, int n_in,
                              void* d_out, int out_size, void* d_ws, size_t ws_size,
                              hipStream_t stream) {
    const float* visual = (const float*)d_in[0];  // [B,V]
    const float* text   = (const float*)d_in[1];  // [B,T]
    const float* w_vis  = (const float*)d_in[2];  // [T]
    const float* w_text = (const float*)d_in[3];  // [V]
    const float* bias   = (const float*)d_in[4];  // [V]
    const float* W_fv   = (const float*)d_in[5];  // [C,T]
    const float* b_fv   = (const float*)d_in[6];  // [C]
    const float* W_ft   = (const float*)d_in[7];  // [C,V]
    const float* b_ft   = (const float*)d_in[8];  // [C]
    float* out = (float*)d_out;                   // [B,C]

    float* ws           = (float*)d_ws;
    float* text_score   = ws;                     // [B,V]
    float* vscaled      = ws + BDIM * VDIM;       // [B,V]
    float* visual_score = ws + 2 * BDIM * VDIM;   // [B,T]

    k_colpass<<<dim3(VDIM / 256, BDIM), dim3(256), 0, stream>>>(
        visual, text, w_vis, w_text, bias, text_score, vscaled);
    k_rowpass<<<dim3(TDIM / 256, BDIM), dim3(256), 0, stream>>>(
        visual, text, w_vis, w_text, bias, vscaled, visual_score);
    k_fusion_wmma<<<dim3(CDIM / 64, BDIM / 16), dim3(128), 0, stream>>>(
        visual_score, text_score, W_fv, b_fv, W_ft, b_ft, out);
}